// TemporalAttentionLayer_27212912787960
// MI455X (gfx1250) — compile-verified
//
#include <hip/hip_runtime.h>
#include <hip/hip_bf16.h>
#include <math.h>

// Problem constants (from reference)
#define BATCH 64
#define KN    100      // window size / num nodes
#define FD    128      // features
#define ED    200      // embed dim (2*K)
#define ALPHA 0.2f

typedef __attribute__((ext_vector_type(2))) float v2f;
typedef __attribute__((ext_vector_type(8))) float v8f;

// ---------------------------------------------------------------------------
// CDNA5 async memory->LDS helpers (GLOBAL_LOAD_ASYNC_TO_LDS_*, ASYNCcnt).
// vdst = LDS byte address (wave-relative, = low 32 bits of generic pointer),
// vaddr = 64-bit global address, saddr = off.
// ---------------------------------------------------------------------------
__device__ __forceinline__ void async_ld_b32(unsigned lds_addr, const void* g) {
    asm volatile("global_load_async_to_lds_b32 %0, %1, off"
                 :: "v"(lds_addr), "v"((unsigned long long)(uintptr_t)g) : "memory");
}
__device__ __forceinline__ void async_ld_b128(unsigned lds_addr, const void* g) {
    asm volatile("global_load_async_to_lds_b128 %0, %1, off"
                 :: "v"(lds_addr), "v"((unsigned long long)(uintptr_t)g) : "memory");
}
__device__ __forceinline__ void wait_async0() {
    asm volatile("s_wait_asynccnt 0" ::: "memory");
}

// ---------------------------------------------------------------------------
// Kernel A: fused projection GEMM via f32 WMMA.
//   Lp[r,e] = x[r,:] @ W[0:128, e] + bvec[e]   (bias folded)
//   Rp[r,e] = x[r,:] @ W[128:256, e]
//   One (6400 x 128) @ (128 x 400) GEMM; cols [0,200)->Lp, [200,400)->Rp.
//   One wave32 = one 16x16 tile, v_wmma_f32_16x16x4_f32 x 32 k-steps.
// ---------------------------------------------------------------------------
__global__ __launch_bounds__(128) void proj_wmma_kernel(
    const float* __restrict__ x,     // (6400, 128)
    const float* __restrict__ W,     // (256, 200)
    const float* __restrict__ bvec,  // (200,)
    float* __restrict__ Lp,          // (6400, 200)
    float* __restrict__ Rp)          // (6400, 200)
{
    const int wave = threadIdx.x >> 5;
    const int lane = threadIdx.x & 31;
    const int tile = blockIdx.x * 4 + wave;          // 0..9999, exact
    const int tm = tile / 25;                        // M tile (0..399)
    const int tn = tile - tm * 25;                   // N tile (0..24)

    const int l16   = lane & 15;
    const int hlf   = lane >> 4;                     // 0 or 1
    const int khalf = hlf * 2;                       // K sub-offset 0 or 2

    // A 16x4 f32 layout: lanes 0-15 = M rows; VGPR0 = K{0,2}, VGPR1 = K{1,3}
    const int mrow = tm * 16 + l16;
    // B 4x16 layout: lanes = N cols; VGPR0 = K row {0,2}, VGPR1 = K row {1,3}
    const int ncol = tn * 16 + l16;
    const bool isL  = (ncol < ED);
    const int wcol  = isL ? ncol : (ncol - ED);
    const int wrow0 = isL ? 0 : FD;

    const float* __restrict__ xrow = x + (size_t)mrow * FD;
    const float* __restrict__ wcolp = W + (size_t)wrow0 * ED + wcol;

    v8f c = {};
    #pragma unroll 8
    for (int k0 = 0; k0 < FD; k0 += 4) {
        const int ka = k0 + khalf;
        v2f afrag, bfrag;
        afrag.x = xrow[ka];
        afrag.y = xrow[ka + 1];
        bfrag.x = wcolp[(size_t)ka * ED];
        bfrag.y = wcolp[(size_t)(ka + 1) * ED];
        // (neg_a, A, neg_b, B, c_mod, C, reuse_a, reuse_b)
        c = __builtin_amdgcn_wmma_f32_16x16x4_f32(false, afrag, false, bfrag,
                                                  (short)0, c, false, false);
    }

    // C/D layout: VGPR v -> M = tm*16 + v + hlf*8 ; N = lane&15.
    // Branch-free epilogue: per-lane constant dest base + bias addend.
    float addv = 0.0f;
    if (isL) addv = bvec[wcol];
    const int baseRow = tm * 16 + hlf * 8;
    float* __restrict__ dst =
        (isL ? Lp : Rp) + (size_t)baseRow * ED + wcol;
    #pragma unroll
    for (int v = 0; v < 8; ++v)
        dst[(size_t)v * ED] = c[v] + addv;
}

// ---------------------------------------------------------------------------
// Kernel B: fused pairwise LeakyReLU attention + softmax + attn@x + sigmoid.
//   One workgroup = (batch b, 25 rows i).  LDS staged with async loads:
//   Rp[b] (padded, stride 201 words -> bank-conflict-free), x[b], Lp slice, a.
//   8 wave32s; waves own rows i, lanes own j in {lane, +32, +64, +96}.
// ---------------------------------------------------------------------------
#define SPLIT       4
#define ROWS_PER_WG (KN / SPLIT)                 // 25
#define RP_STRIDE   201                          // word stride: 201*4B -> bank step 9 (gcd(9,64)=1)
#define RP_WORDS    (KN * RP_STRIDE)             // 20100
#define X_WORDS     (KN * FD)                    // 12800
#define LP_WORDS    (ROWS_PER_WG * ED)           // 5000
#define A_WORDS     ED                           // 200
#define ATT_WORDS   (8 * 128)                    // per-wave softmax rows
#define SMEM_WORDS  (RP_WORDS + X_WORDS + LP_WORDS + A_WORDS + ATT_WORDS)
#define SMEM_BYTES  (SMEM_WORDS * 4)

__global__ __launch_bounds__(256) void attn_fused_kernel(
    const float* __restrict__ Lp,    // (6400, 200), bias-folded
    const float* __restrict__ Rp,    // (6400, 200)
    const float* __restrict__ x,     // (B, K, F)
    const float* __restrict__ avec,  // (200,)
    const float* __restrict__ bias,  // (K, K)
    float* __restrict__ out)         // (B, K, F)
{
    extern __shared__ float smem[];
    float* Rp_s  = smem;                          // [100][201]
    float* x_s   = Rp_s + RP_WORDS;               // [100][128]
    float* Lp_s  = x_s + X_WORDS;                 // [25][200]
    float* a_s   = Lp_s + LP_WORDS;               // [200]
    float* att_s = a_s + A_WORDS;                 // [8][128]

    const unsigned lds0 = (unsigned)(uintptr_t)smem;   // wave-relative LDS byte base
    const unsigned rp_b  = lds0;
    const unsigned x_b   = lds0 + RP_WORDS * 4u;
    const unsigned lp_b  = x_b + X_WORDS * 4u;
    const unsigned a_b   = lp_b + LP_WORDS * 4u;

    const int b    = blockIdx.x / SPLIT;
    const int part = blockIdx.x - b * SPLIT;
    const int i0   = part * ROWS_PER_WG;
    const int tid  = threadIdx.x;
    const int wave = tid >> 5;
    const int lane = tid & 31;

    const float* __restrict__ Rp_g = Rp + (size_t)b * KN * ED;
    const float* __restrict__ Lp_g = Lp + (size_t)b * KN * ED + (size_t)i0 * ED;
    const float* __restrict__ x_g  = x  + (size_t)b * KN * FD;

    // ---- async stage batch data into LDS (GLOBAL_LOAD_ASYNC_TO_LDS_*) ----
    // Rp: padded rows (4B-aligned only) -> b32 per element
    for (int idx = tid; idx < KN * ED; idx += 256) {
        const int j = idx / ED;
        const int e = idx - j * ED;
        async_ld_b32(rp_b + (unsigned)(j * RP_STRIDE + e) * 4u, Rp_g + idx);
    }
    // x, Lp slice, a: contiguous + 16B aligned -> b128
    for (int t = tid; t < X_WORDS / 4; t += 256)
        async_ld_b128(x_b + (unsigned)t * 16u, x_g + t * 4);
    for (int t = tid; t < LP_WORDS / 4; t += 256)
        async_ld_b128(lp_b + (unsigned)t * 16u, Lp_g + t * 4);
    for (int t = tid; t < A_WORDS / 4; t += 256)
        async_ld_b128(a_b + (unsigned)t * 16u, avec + t * 4);
    wait_async0();          // drain this wave's ASYNCcnt
    __syncthreads();        // then all waves' LDS writes are visible

    // j columns owned by this lane
    const int j0 = lane;
    const int j1 = lane + 32;
    const int j2 = lane + 64;
    const bool j3ok = (lane < (KN - 96));         // lane < 4
    const int j3 = j3ok ? (lane + 96) : 0;

    const float* __restrict__ R0 = Rp_s + j0 * RP_STRIDE;
    const float* __restrict__ R1 = Rp_s + j1 * RP_STRIDE;
    const float* __restrict__ R2 = Rp_s + j2 * RP_STRIDE;
    const float* __restrict__ R3 = Rp_s + j3 * RP_STRIDE;

    float* __restrict__ att = att_s + wave * 128;

    for (int i = i0 + wave; i < i0 + ROWS_PER_WG; i += 8) {
        const float* __restrict__ Lrow = Lp_s + (i - i0) * ED;

        // ---- e[i,j] = sum_e leaky(Lp+Rp) * a ; leaky(p) = max(p, 0.2p) ----
        float acc0 = 0.f, acc1 = 0.f, acc2 = 0.f, acc3 = 0.f;
        #pragma unroll 2
        for (int e = 0; e < ED; ++e) {
            const float lv = Lrow[e];             // LDS broadcast
            const float av = a_s[e];              // LDS broadcast
            float p;
            p = lv + R0[e]; acc0 = __builtin_fmaf(fmaxf(p, ALPHA * p), av, acc0);
            p = lv + R1[e]; acc1 = __builtin_fmaf(fmaxf(p, ALPHA * p), av, acc1);
            p = lv + R2[e]; acc2 = __builtin_fmaf(fmaxf(p, ALPHA * p), av, acc2);
            p = lv + R3[e]; acc3 = __builtin_fmaf(fmaxf(p, ALPHA * p), av, acc3);
        }
        const float* brow = bias + (size_t)i * KN;
        float e0 = acc0 + brow[j0];
        float e1 = acc1 + brow[j1];
        float e2 = acc2 + brow[j2];
        float e3 = j3ok ? (acc3 + brow[j3]) : -INFINITY;

        // ---- wave32 softmax over 100 j values ----
        float m = fmaxf(fmaxf(e0, e1), fmaxf(e2, e3));
        #pragma unroll
        for (int off = 16; off > 0; off >>= 1)
            m = fmaxf(m, __shfl_xor(m, off, 32));
        float p0 = __expf(e0 - m);
        float p1 = __expf(e1 - m);
        float p2 = __expf(e2 - m);
        float p3 = j3ok ? __expf(e3 - m) : 0.f;
        float s = p0 + p1 + p2 + p3;
        #pragma unroll
        for (int off = 16; off > 0; off >>= 1)
            s += __shfl_xor(s, off, 32);
        const float inv = 1.0f / s;

        att[j0] = p0 * inv;
        att[j1] = p1 * inv;
        att[j2] = p2 * inv;
        if (j3ok) att[j3] = p3 * inv;
        asm volatile("s_wait_dscnt 0" ::: "memory");   // wave-local LDS store->load

        // ---- h[i,f] = sigmoid( sum_j att[j] * x[j,f] ) ; lanes own f ----
        float h0 = 0.f, h1 = 0.f, h2 = 0.f, h3 = 0.f;
        #pragma unroll 2
        for (int j = 0; j < KN; ++j) {
            const float w = att[j];               // LDS broadcast
            const float* xr = x_s + j * FD + lane;
            h0 = __builtin_fmaf(w, xr[0],  h0);
            h1 = __builtin_fmaf(w, xr[32], h1);
            h2 = __builtin_fmaf(w, xr[64], h2);
            h3 = __builtin_fmaf(w, xr[96], h3);
        }
        float* orow = out + (size_t)b * KN * FD + (size_t)i * FD + lane;
        orow[0]  = 1.0f / (1.0f + __expf(-h0));
        orow[32] = 1.0f / (1.0f + __expf(-h1));
        orow[64] = 1.0f / (1.0f + __expf(-h2));
        orow[96] = 1.0f / (1.0f + __expf(-h3));
    }
}

// ---------------------------------------------------------------------------
// Host launch
// ---------------------------------------------------------------------------
extern "C" void kernel_launch(void* const* d_in, const int* in_sizes, int n_in,
                              void* d_out, int out_size, void* d_ws, size_t ws_size,
                              hipStream_t stream) {
    const float* x    = (const float*)d_in[0];   // (64,100,128)
    const float* W    = (const float*)d_in[1];   // (256,200)
    const float* bvec = (const float*)d_in[2];   // (200,)
    const float* avec = (const float*)d_in[3];   // (200,)
    const float* bias = (const float*)d_in[4];   // (100,100)
    float* out = (float*)d_out;

    float* Lp = (float*)d_ws;                            // 6400*200 f32 = 5.12 MB
    float* Rp = Lp + (size_t)BATCH * KN * ED;            // 6400*200 f32 = 5.12 MB

    (void)hipFuncSetAttribute((const void*)attn_fused_kernel,
                              hipFuncAttributeMaxDynamicSharedMemorySize, SMEM_BYTES);

    // 10000 WMMA tiles, 4 waves (tiles) per 128-thread block
    proj_wmma_kernel<<<2500, 128, 0, stream>>>(x, W, bvec, Lp, Rp);

    // one WG per (batch, 25-row slice): 64*4 = 256 WGs, 8 wave32s each
    attn_fused_kernel<<<BATCH * SPLIT, 256, SMEM_BYTES, stream>>>(Lp, Rp, x, avec, bias, out);
}